// TurbulentKineticEnergySpectrum_56556129354598
// MI455X (gfx1250) — compile-verified
//
#include <hip/hip_runtime.h>
#include <math.h>

// ---------------------------------------------------------------------------
// Types for WMMA f32 16x16x4: A is 16x4 f32 (2 VGPRs), B is 4x16 f32 (2 VGPRs),
// C/D is 16x16 f32 (8 VGPRs). Wave32.
// ---------------------------------------------------------------------------
typedef float v2f __attribute__((ext_vector_type(2)));
typedef float v8f __attribute__((ext_vector_type(8)));

static __device__ __forceinline__ v8f wmma_k4(v2f a, v2f b, v8f c) {
  // 8 args: (neg_a, A, neg_b, B, c_mod, C, reuse_a, reuse_b)
  return __builtin_amdgcn_wmma_f32_16x16x4_f32(false, a, false, b, (short)0, c,
                                               false, false);
}

// ---------------------------------------------------------------------------
// Kernel 1: DFT-128 twiddle matrix, W[m][n] = exp(-2*pi*i*m*n/128)
// stored as separate re / im planes (row-major [m][n]).
// ---------------------------------------------------------------------------
__global__ __launch_bounds__(256) void dft_init_kernel(float* __restrict__ dre,
                                                       float* __restrict__ dimg) {
  int idx = blockIdx.x * 256 + threadIdx.x;
  if (idx >= 128 * 128) return;
  int m = idx >> 7, n = idx & 127;
  int r = (m * n) & 127;                       // exploit periodicity for accuracy
  float a = -6.283185307179586f * (float)r * (1.0f / 128.0f);
  dre[idx]  = cosf(a);
  dimg[idx] = sinf(a);
}

// ---------------------------------------------------------------------------
// Kernel 2: tke = 0.5 * sum_c u^2, written as complex (im = 0).
// u layout (B, 3, X, Y, Z); buf layout (B, X, Y, Z) of float2.
// ---------------------------------------------------------------------------
__global__ __launch_bounds__(256) void tke_kernel(const float* __restrict__ u,
                                                  float2* __restrict__ buf,
                                                  int total) {
  int idx = blockIdx.x * 256 + threadIdx.x;
  if (idx >= total) return;
  int b = idx >> 21;                 // / 2097152
  int r = idx & 2097151;
  const float* ub = u + (size_t)b * 3u * 2097152u + r;
  float a0 = ub[0];
  float a1 = ub[2097152];
  float a2 = ub[2 * 2097152];
  float t = 0.5f * (a0 * a0 + a1 * a1 + a2 * a2);
  float2 o; o.x = t; o.y = 0.0f;
  buf[idx] = o;
}

// ---------------------------------------------------------------------------
// Kernel 3: one in-place 1-D DFT pass (length 128) along a strided dimension,
// done as a complex GEMM  F(128 x 16lines) = W(128x128) * x(128 x 16lines)
// with f32 WMMA 16x16x4 accumulation chains.
//
// Line base address: base(l) = (l / lpo) * outerStride + (l % lpo) * lineStride
// Element t of a line lives at base + t * tStride.   (all in float2 elements)
//
// Block = 128 threads (4 waves); block handles 16 lines; each wave produces
// two 16-row frequency tiles for all 16 lines in a single fused chunk loop
// (8 independent WMMA accumulator chains, B operands loaded once from LDS).
//
// LDS is SoA (re/im planes) with rows padded to 132 floats so the 16 column
// readers (lane&15) land on distinct banks (stride 132 mod 64 = 4).
// ---------------------------------------------------------------------------
__global__ __launch_bounds__(128) void fft_pass_kernel(
    float2* __restrict__ buf, const float* __restrict__ dre,
    const float* __restrict__ dimg, int lpo, int lineStride, int outerStride,
    int tStride) {
  __shared__ float tile_re[16][132];
  __shared__ float tile_im[16][132];

  const int tid = threadIdx.x;
  const int l0 = blockIdx.x * 16;

  // Stage 16 lines x 128 samples into LDS (SoA).
  for (int i = tid; i < 16 * 128; i += 128) {
    int ll = i >> 7;
    int t  = i & 127;
    int gl = l0 + ll;
    size_t base = (size_t)(gl / lpo) * (size_t)outerStride +
                  (size_t)(gl % lpo) * (size_t)lineStride;
    float2 v = buf[base + (size_t)t * (size_t)tStride];
    tile_re[ll][t] = v.x;
    tile_im[ll][t] = v.y;
  }
  __syncthreads();

  const int lane = tid & 31;
  const int wave = tid >> 5;
  const int lcol = lane & 15;        // N column (line) for B/C/D operands
  const int hi   = lane >> 4;        // 0 | 1 (upper half-wave)
  const int koff = hi << 1;          // K offset per ISA A/B layout (0 | 2)

  const int gline = l0 + lcol;
  const size_t lineBase = (size_t)(gline / lpo) * (size_t)outerStride +
                          (size_t)(gline % lpo) * (size_t)lineStride;

  // Two 16-row output tiles per wave, fused into one K loop.
  const int m0 = (wave * 2) * 16 + lcol;           // A row for tile 0
  const float* __restrict__ ar0 = dre  + m0 * 128;
  const float* __restrict__ ai0 = dimg + m0 * 128;
  const float* __restrict__ ar1 = ar0 + 16 * 128;  // tile 1 = rows m0+16
  const float* __restrict__ ai1 = ai0 + 16 * 128;

  v8f aRR0 = {}, aII0 = {}, aRI0 = {}, aIR0 = {};
  v8f aRR1 = {}, aII1 = {}, aRI1 = {}, aIR1 = {};

#pragma unroll 4
  for (int c = 0; c < 32; ++c) {
    const int n0 = (c << 2) + koff;                // even -> 8B aligned
    v2f b_re = *(const v2f*)&tile_re[lcol][n0];    // one ds_load_b64 each
    v2f b_im = *(const v2f*)&tile_im[lcol][n0];
    v2f a_re0 = *(const v2f*)(ar0 + n0);
    v2f a_im0 = *(const v2f*)(ai0 + n0);
    v2f a_re1 = *(const v2f*)(ar1 + n0);
    v2f a_im1 = *(const v2f*)(ai1 + n0);
    aRR0 = wmma_k4(a_re0, b_re, aRR0);
    aII0 = wmma_k4(a_im0, b_im, aII0);
    aRI0 = wmma_k4(a_re0, b_im, aRI0);
    aIR0 = wmma_k4(a_im0, b_re, aIR0);
    aRR1 = wmma_k4(a_re1, b_re, aRR1);
    aII1 = wmma_k4(a_im1, b_im, aII1);
    aRI1 = wmma_k4(a_re1, b_im, aRI1);
    aIR1 = wmma_k4(a_im1, b_re, aIR1);
  }

  // C/D layout: lane 0-15 -> M = v, lane 16-31 -> M = v + 8; N = lane & 15.
#pragma unroll
  for (int v = 0; v < 8; ++v) {
    int freq0 = (wave * 2) * 16 + v + (hi << 3);
    float2 o0;
    o0.x = aRR0[v] - aII0[v];
    o0.y = aRI0[v] + aIR0[v];
    buf[lineBase + (size_t)freq0 * (size_t)tStride] = o0;
    int freq1 = freq0 + 16;
    float2 o1;
    o1.x = aRR1[v] - aII1[v];
    o1.y = aRI1[v] + aIR1[v];
    buf[lineBase + (size_t)freq1 * (size_t)tStride] = o1;
  }
}

// ---------------------------------------------------------------------------
// Kernel 4: trilinear interpolation of log|F|^2 (fftshift folded into the
// gather index) + exp + Lebedev quadrature reduction.
// One 256-thread block per (k-shell, batch); output E_k[b][kk].
// ---------------------------------------------------------------------------
static __device__ __forceinline__ float logpow(const float2* __restrict__ Fb,
                                               int x, int y, int z) {
  int xs = (x + 64) & 127;           // fftshift: shifted[i] = F[(i+64)%128]
  int ys = (y + 64) & 127;
  int zs = (z + 64) & 127;
  float2 c = Fb[((size_t)xs << 14) + (ys << 7) + zs];
  float m2 = c.x * c.x + c.y * c.y;
  return logf(fmaxf(m2, 1e-37f));
}

__global__ __launch_bounds__(256) void interp_kernel(
    const float2* __restrict__ F, const float* __restrict__ kArr,
    const float* __restrict__ p, const float* __restrict__ w,
    float* __restrict__ out, int nK, int nLeb) {
  __shared__ float red[256];
  const int kk = blockIdx.x;
  const int b  = blockIdx.y;
  const float kv = kArr[kk];
  const float2* __restrict__ Fb = F + (size_t)b * 2097152u;

  float acc = 0.0f;
  for (int n = threadIdx.x; n < nLeb; n += 256) {
    float qx = fmaf(kv, p[3 * n + 0], 64.0f);
    float qy = fmaf(kv, p[3 * n + 1], 64.0f);
    float qz = fmaf(kv, p[3 * n + 2], 64.0f);
    int fx = (int)floorf(qx), fy = (int)floorf(qy), fz = (int)floorf(qz);
    int x0 = min(max(fx, 0), 127), x1 = min(max(fx + 1, 0), 127);
    int y0 = min(max(fy, 0), 127), y1 = min(max(fy + 1, 0), 127);
    int z0 = min(max(fz, 0), 127), z1 = min(max(fz + 1, 0), 127);
    float wx = qx - (float)x0;       // matches reference (subtract clipped p0)
    float wy = qy - (float)y0;
    float wz = qz - (float)z0;

    float c000 = logpow(Fb, x0, y0, z0), c001 = logpow(Fb, x0, y0, z1);
    float c010 = logpow(Fb, x0, y1, z0), c011 = logpow(Fb, x0, y1, z1);
    float c100 = logpow(Fb, x1, y0, z0), c101 = logpow(Fb, x1, y0, z1);
    float c110 = logpow(Fb, x1, y1, z0), c111 = logpow(Fb, x1, y1, z1);

    float v = (1.0f - wx) * ((1.0f - wy) * ((1.0f - wz) * c000 + wz * c001) +
                             wy * ((1.0f - wz) * c010 + wz * c011)) +
              wx * ((1.0f - wy) * ((1.0f - wz) * c100 + wz * c101) +
                    wy * ((1.0f - wz) * c110 + wz * c111));
    acc += expf(v) * w[n];
  }

  red[threadIdx.x] = acc;
  __syncthreads();
  for (int s = 128; s > 0; s >>= 1) {
    if (threadIdx.x < s) red[threadIdx.x] += red[threadIdx.x + s];
    __syncthreads();
  }
  if (threadIdx.x == 0)
    out[b * nK + kk] = red[0] * 12.566370614359172f * kv * kv;   // 4*pi*k^2
}

// ---------------------------------------------------------------------------
// Host launcher.
// Inputs: [0] u (B,3,128,128,128) f32   [1] k (K,) f32
//         [2] p (N,3) f32               [3] w (N,) f32
// Output: E_k (B,K) f32.
// Workspace: [0, 64MiB) complex grid buffer; then DFT re/im planes (128KB).
// ---------------------------------------------------------------------------
extern "C" void kernel_launch(void* const* d_in, const int* in_sizes, int n_in,
                              void* d_out, int out_size, void* d_ws,
                              size_t ws_size, hipStream_t stream) {
  const float* u = (const float*)d_in[0];
  const float* k = (const float*)d_in[1];
  const float* p = (const float*)d_in[2];
  const float* w = (const float*)d_in[3];
  float* out = (float*)d_out;

  const int nK = in_sizes[1];
  const int nLeb = in_sizes[3];
  const int B = in_sizes[0] / (3 * 2097152);     // 4

  char* ws = (char*)d_ws;
  float2* buf  = (float2*)ws;                            // B*128^3 float2
  size_t bufBytes = (size_t)B * 2097152u * sizeof(float2);
  float* dre  = (float*)(ws + bufBytes);                 // 64 KB
  float* dimg = (float*)(ws + bufBytes + 65536);         // 64 KB

  // 1. DFT twiddle table
  dft_init_kernel<<<64, 256, 0, stream>>>(dre, dimg);

  // 2. tke -> complex grid
  const int total = B * 2097152;
  tke_kernel<<<(total + 255) / 256, 256, 0, stream>>>(u, buf, total);

  // 3. Three in-place DFT passes (Z, then Y, then X).
  const int blocksPerPass = (B * 16384) / 16;
  // Z: lines over (B,X,Y); base = line*128; stride along transform = 1
  fft_pass_kernel<<<blocksPerPass, 128, 0, stream>>>(buf, dre, dimg,
                                                     B * 16384, 128, 0, 1);
  // Y: lines over (B,X) x Z; base = outer*16384 + z; stride = 128
  fft_pass_kernel<<<blocksPerPass, 128, 0, stream>>>(buf, dre, dimg,
                                                     128, 1, 16384, 128);
  // X: lines over B x (Y,Z); base = b*2097152 + yz; stride = 16384
  fft_pass_kernel<<<blocksPerPass, 128, 0, stream>>>(buf, dre, dimg,
                                                     16384, 1, 2097152, 16384);

  // 4. Interpolate + quadrature
  interp_kernel<<<dim3(nK, B), 256, 0, stream>>>(buf, k, p, w, out, nK, nLeb);
}